// WaveConv2d_65738769433248
// MI455X (gfx1250) — compile-verified
//
#include <hip/hip_runtime.h>

// WaveConv2d for MI455X (gfx1250, wave32).
// Fused LDS-tiled 4-level db4 DWT -> per-position channel GEMM on level-4
// subbands via V_WMMA_F32_16X16X4_F32 (fully unrolled K) -> fused 4-level
// inverse DWT. All scratch in d_ws, all launches on `stream`.

#define BC_ 1024   // B*C = 16*64
#define CCH 64     // channels
#define NB  16     // batch

// db4 decomposition filters (DEC_LO / DEC_HI). Analysis uses reversed taps
// (index 7-k); synthesis uses them directly (lax.conv = cross-correlation).
__constant__ float c_dlo[8] = {
  -0.010597401784997278f,  0.032883011666982945f,  0.030841381835986965f,
  -0.18703481171888114f,  -0.02798376941698385f,   0.6308807679295904f,
   0.7148465705525415f,    0.23037781330885523f };
__constant__ float c_dhi[8] = {
  -0.23037781330885523f,   0.7148465705525415f,   -0.6308807679295904f,
  -0.02798376941698385f,   0.18703481171888114f,   0.030841381835986965f,
  -0.032883011666982945f, -0.010597401784997278f };

static __device__ __forceinline__ int symfold(int m, int N) {
  if (m < 0)  m = -m - 1;
  if (m >= N) m = 2 * N - 1 - m;
  return m;
}

// ============================================================================
// Fused one-level 2D analysis. in: (BC,H,W) plane with channel stride inCS.
// out: (BC,4,Ho,Wo), subband s = fh*2+fv (s0=LL, s1..s3 = reference highs
// order). 16x16 output tile per block; 38x38 input tile staged in LDS,
// horizontal taps -> LDS, vertical taps -> 4 subbands. pad-left = 6 always.
// ============================================================================
__global__ __launch_bounds__(256) void afb2d_fused(
    const float* __restrict__ in, long inCS,
    float* __restrict__ out, int H, int W, int Ho, int Wo) {
  __shared__ float sIn[38][39];     // +1 col pad vs bank conflicts
  __shared__ float sLo[38][16];
  __shared__ float sHi[38][16];

  int bc = blockIdx.z;
  int woBase = blockIdx.x * 16;
  int hoBase = blockIdx.y * 16;
  int tid = threadIdx.x;
  const float* src = in + (long)bc * inCS;
  int imBase = 2 * hoBase - 6;
  int icBase = 2 * woBase - 6;

  // stage 1: input tile (symmetric extension folded at load)
  for (int idx = tid; idx < 38 * 38; idx += 256) {
    int r = idx / 38, c = idx % 38;
    int gr = symfold(imBase + r, H);
    int gc = symfold(icBase + c, W);
    sIn[r][c] = src[(long)gr * W + gc];
  }
  __syncthreads();

  // stage 2: horizontal analysis -> intermediate lo/hi rows
  for (int idx = tid; idx < 38 * 16; idx += 256) {
    int r = idx / 16, c = idx % 16;
    float lo = 0.f, hi = 0.f;
#pragma unroll
    for (int k = 0; k < 8; ++k) {
      float v = sIn[r][2 * c + k];
      lo = fmaf(v, c_dlo[7 - k], lo);
      hi = fmaf(v, c_dhi[7 - k], hi);
    }
    sLo[r][c] = lo;
    sHi[r][c] = hi;
  }
  __syncthreads();

  // stage 3: vertical analysis -> LL, LH, HL, HH
  int wol = tid & 15, hol = tid >> 4;
  int wo = woBase + wol, ho = hoBase + hol;
  if (wo < Wo && ho < Ho) {
    float a00 = 0.f, a01 = 0.f, a10 = 0.f, a11 = 0.f;
#pragma unroll
    for (int k = 0; k < 8; ++k) {
      float vl = sLo[2 * hol + k][wol];
      float vh = sHi[2 * hol + k][wol];
      float fl = c_dlo[7 - k], fh = c_dhi[7 - k];
      a00 = fmaf(vl, fl, a00);   // fh=0, fv=0 : LL
      a01 = fmaf(vl, fh, a01);   // fh=0, fv=1 : s1
      a10 = fmaf(vh, fl, a10);   // fh=1, fv=0 : s2
      a11 = fmaf(vh, fh, a11);   // fh=1, fv=1 : s3
    }
    long ps = (long)Ho * Wo;
    long ob = ((long)bc * 4 * Ho + ho) * Wo + wo;
    out[ob]          = a00;
    out[ob + ps]     = a01;
    out[ob + 2 * ps] = a10;
    out[ob + 3 * ps] = a11;
  }
}

// ============================================================================
// Spectral channel mix:  out[b,o,s,p] = sum_i a[b,i,s,p] * w_s[i,o,p]
// One wave per (p, s, ntile): M=16 batches, N=16-wide o tile, K=64 input
// channels as 16 fully-unrolled V_WMMA_F32_16X16X4_F32 steps.
// ============================================================================
typedef __attribute__((ext_vector_type(2))) float v2f;
typedef __attribute__((ext_vector_type(8))) float v8f;

__global__ __launch_bounds__(128) void mul2d_wmma(
    const float* __restrict__ A,   // (B,C,4,484)
    const float* __restrict__ W1, const float* __restrict__ W2,
    const float* __restrict__ W3, const float* __restrict__ W4,
    float* __restrict__ Dst) {     // (B,C,4,484)
  const int HW = 484;
  int p = blockIdx.x;
  int s = blockIdx.y;
  int ntile = threadIdx.x >> 5;   // 4 waves/block = 4 o-tiles
  int lane  = threadIdx.x & 31;
  const float* Wm = (s == 0) ? W1 : (s == 1) ? W2 : (s == 2) ? W3 : W4;

  int half = lane >> 4;           // 0: lanes 0-15, 1: lanes 16-31
  int mr   = lane & 15;           // m for A, n for B
  int obase = ntile * 16;
  v8f acc = {};

#pragma unroll
  for (int k = 0; k < 64; k += 4) {
    int ka = k + half * 2;        // A VGPR0: K={0,2}+k, VGPR1: K={1,3}+k
    v2f a, b;
    a.x = A[(((long)mr * CCH + (ka + 0)) * 4 + s) * HW + p];
    a.y = A[(((long)mr * CCH + (ka + 1)) * 4 + s) * HW + p];
    b.x = Wm[((long)(ka + 0) * CCH + obase + mr) * HW + p];
    b.y = Wm[((long)(ka + 1) * CCH + obase + mr) * HW + p];
    acc = __builtin_amdgcn_wmma_f32_16x16x4_f32(
        false, a, false, b, (short)0, acc, false, false);
  }
#pragma unroll
  for (int r = 0; r < 8; ++r) {
    int m = r + half * 8;
    int o = obase + mr;
    Dst[(((long)m * CCH + o) * 4 + s) * HW + p] = acc[r];
  }
}

// ============================================================================
// Fused one-level 2D synthesis (lhs_dilation=2, pad=1, zero OOB).
// ll: (BC,Hin,Win) w/ channel stride llCS. hb: s=1 high plane, plane stride
// hPS, channel stride hCS (s=2,3 at +hPS,+2*hPS). out: (BC,Hout,Wout),
// Hout/Wout already cropped. 32x32 output tile; 20x20 tile per subband.
// ============================================================================
__global__ __launch_bounds__(256) void sfb2d_fused(
    const float* __restrict__ ll, long llCS,
    const float* __restrict__ hb, long hCS, long hPS,
    float* __restrict__ out, int Hin, int Win, int Hout, int Wout) {
  __shared__ float sB[4][20][20];
  __shared__ float sLoT[32][20];
  __shared__ float sHiT[32][20];

  int bc = blockIdx.z;
  int xBase = blockIdx.x * 32, yBase = blockIdx.y * 32;
  int tBase = yBase >> 1, cBase = xBase >> 1;
  int tid = threadIdx.x;
  const float* p0 = ll + (long)bc * llCS;
  const float* p1 = hb + (long)bc * hCS;
  const float* p2 = p1 + hPS;
  const float* p3 = p2 + hPS;

  // stage 1: 4 subband tiles (zero beyond input extent = conv zero-pad)
  for (int idx = tid; idx < 4 * 400; idx += 256) {
    int sbi = idx / 400;
    int rem = idx % 400;
    int r = rem / 20, c = rem % 20;
    int t = tBase + r, cc = cBase + c;
    const float* p = (sbi == 0) ? p0 : (sbi == 1) ? p1 : (sbi == 2) ? p2 : p3;
    float v = 0.f;
    if (t < Hin && cc < Win) v = p[(long)t * Win + cc];
    sB[sbi][r][c] = v;
  }
  __syncthreads();

  // stage 2: vertical synthesis -> loT (LL,s1), hiT (s2,s3)
  for (int idx = tid; idx < 32 * 20; idx += 256) {
    int c = idx % 20, yl = idx / 20;
    int tl = yl >> 1;
    int kb = (yl & 1) ? 0 : 1;
    float lo = 0.f, hi = 0.f;
#pragma unroll
    for (int j = 0; j < 4; ++j) {
      int k = 2 * j + kb;
      lo = fmaf(sB[0][tl + j][c], c_dlo[k], lo);
      lo = fmaf(sB[1][tl + j][c], c_dhi[k], lo);
      hi = fmaf(sB[2][tl + j][c], c_dlo[k], hi);
      hi = fmaf(sB[3][tl + j][c], c_dhi[k], hi);
    }
    sLoT[yl][c] = lo;
    sHiT[yl][c] = hi;
  }
  __syncthreads();

  // stage 3: horizontal synthesis -> output tile (4 rows per thread)
  int xl = tid & 31;
  int yl0 = tid >> 5;
  int x = xBase + xl;
  int kbx = (xl & 1) ? 0 : 1;
  int ttl = xl >> 1;
#pragma unroll
  for (int q = 0; q < 4; ++q) {
    int yl = yl0 + 8 * q;
    int y = yBase + yl;
    if (x < Wout && y < Hout) {
      float acc = 0.f;
#pragma unroll
      for (int j = 0; j < 4; ++j) {
        int k = 2 * j + kbx;
        acc = fmaf(sLoT[yl][ttl + j], c_dlo[k], acc);
        acc = fmaf(sHiT[yl][ttl + j], c_dhi[k], acc);
      }
      out[((long)bc * Hout + y) * Wout + x] = acc;
    }
  }
}

static inline int cdivi(int a, int b) { return (a + b - 1) / b; }

extern "C" void kernel_launch(void* const* d_in, const int* in_sizes, int n_in,
                              void* d_out, int out_size, void* d_ws, size_t ws_size,
                              hipStream_t stream) {
  (void)in_sizes; (void)n_in; (void)out_size; (void)ws_size;
  const float* x  = (const float*)d_in[0];
  const float* w1 = (const float*)d_in[1];
  const float* w2 = (const float*)d_in[2];
  const float* w3 = (const float*)d_in[3];
  const float* w4 = (const float*)d_in[4];
  float* outp = (float*)d_out;

  const long BC = BC_;
  // level dims: 256 -> 131 -> 69 -> 38 -> 22
  const long hw1 = 131L * 131, hw2 = 69L * 69, hw3 = 38L * 38, hw4 = 22L * 22;

  float* ws = (float*)d_ws;
  size_t off = 0;
  float* H1  = ws + off; off += (size_t)BC * 4 * hw1;
  float* H2  = ws + off; off += (size_t)BC * 4 * hw2;
  float* H3  = ws + off; off += (size_t)BC * 4 * hw3;
  float* H4  = ws + off; off += (size_t)BC * 4 * hw4;
  float* NH4 = ws + off; off += (size_t)BC * 4 * hw4;
  float* ll3 = ws + off; off += (size_t)BC * hw3;
  float* ll2 = ws + off; off += (size_t)BC * hw2;
  float* ll1 = ws + off; off += (size_t)BC * hw1;

  // ---------------- forward DWT (fused per level) ----------------
  afb2d_fused<<<dim3(cdivi(131,16), cdivi(131,16), BC), 256, 0, stream>>>(
      x, 256L * 256, H1, 256, 256, 131, 131);
  afb2d_fused<<<dim3(cdivi(69,16), cdivi(69,16), BC), 256, 0, stream>>>(
      H1, 4L * hw1, H2, 131, 131, 69, 69);
  afb2d_fused<<<dim3(cdivi(38,16), cdivi(38,16), BC), 256, 0, stream>>>(
      H2, 4L * hw2, H3, 69, 69, 38, 38);
  afb2d_fused<<<dim3(cdivi(22,16), cdivi(22,16), BC), 256, 0, stream>>>(
      H3, 4L * hw3, H4, 38, 38, 22, 22);

  // ---------------- spectral channel mix (WMMA) ----------------
  mul2d_wmma<<<dim3(484, 4), 128, 0, stream>>>(H4, w1, w2, w3, w4, NH4);

  // ---------------- inverse DWT (fused per level) ----------------
  // L4: 22 -> 38
  sfb2d_fused<<<dim3(cdivi(38,32), cdivi(38,32), BC), 256, 0, stream>>>(
      NH4, 4L * hw4, NH4 + hw4, 4L * hw4, hw4, ll3, 22, 22, 38, 38);
  // L3: 38 -> 69 (crop 70->69 folded into output size)
  sfb2d_fused<<<dim3(cdivi(69,32), cdivi(69,32), BC), 256, 0, stream>>>(
      ll3, hw3, H3 + hw3, 4L * hw3, hw3, ll2, 38, 38, 69, 69);
  // L2: 69 -> 131 (crop 132->131)
  sfb2d_fused<<<dim3(cdivi(131,32), cdivi(131,32), BC), 256, 0, stream>>>(
      ll2, hw2, H2 + hw2, 4L * hw2, hw2, ll1, 69, 69, 131, 131);
  // L1: 131 -> 256 (final output)
  sfb2d_fused<<<dim3(cdivi(256,32), cdivi(256,32), BC), 256, 0, stream>>>(
      ll1, hw1, H1 + hw1, 4L * hw1, hw1, outp, 131, 131, 256, 256);
}